// DynamicMutiScaleContextBlock_5291399709404
// MI455X (gfx1250) — compile-verified
//
#include <hip/hip_runtime.h>
#include <math.h>

// ---------------------------------------------------------------------------
// DynamicMultiScaleContextBlock for MI455X (gfx1250, wave32).
// Memory-bound pipeline (AI << 1 flop/byte); everything kept in f32 since the
// 192MB L2 holds the whole ~150MB working set. The one real GEMM (DCM joint
// SE: 131072x60 @ 60x7 -> relu/bn -> @7x60 -> sigmoid) is fused into a single
// WMMA kernel using V_WMMA_F32_16X16X4_F32 (exact f32), with all weights
// staged zero-padded in LDS so the WMMA inner loop has no divergence.
// ---------------------------------------------------------------------------

typedef float v2f __attribute__((ext_vector_type(2)));
typedef float v8f __attribute__((ext_vector_type(8)));

#define DEVI __device__ __forceinline__

static constexpr int B      = 32;
static constexpr int C0     = 20, H0 = 128, W0 = 128, HW0 = 16384;
static constexpr int C1     = 40, H1 = 64,  W1p = 64, HW1 = 4096;
static constexpr int CTOT   = 60;           // C0 + C1
static constexpr int MID    = 7;            // 60 // 8
static constexpr int MINI   = 4096;         // 64*64

DEVI float bnscale(float g) { return g * rsqrtf(1.0f + 1e-5f); }
DEVI float sigmoidf(float x) { return 1.0f / (1.0f + __expf(-x)); }

// ---- generic: per-pixel channel dot (1x1 conv to 1 channel) ----------------
__global__ void k_chan_logits(const float* __restrict__ x,   // + channel base
                              const float* __restrict__ w,
                              const float* __restrict__ bias,
                              float* __restrict__ out,
                              int C, int HW, int sampleStride) {
  int t  = blockIdx.x * blockDim.x + threadIdx.x;
  int b  = t / HW, hw = t % HW;
  const float* xb = x + (size_t)b * sampleStride + hw;
  float acc = bias[0];
  for (int c = 0; c < C; ++c) acc += xb[(size_t)c * HW] * w[c];
  out[t] = acc;
}

// ---- in-place row softmax (one block per batch row) ------------------------
__global__ void k_softmax_rows(float* __restrict__ p, int n) {
  __shared__ float red[256];
  float* row = p + (size_t)blockIdx.x * n;
  int tid = threadIdx.x;
  float m = -3.402823e38f;
  for (int i = tid; i < n; i += 256) m = fmaxf(m, row[i]);
  red[tid] = m; __syncthreads();
  for (int s = 128; s > 0; s >>= 1) { if (tid < s) red[tid] = fmaxf(red[tid], red[tid + s]); __syncthreads(); }
  m = red[0]; __syncthreads();
  float sum = 0.f;
  for (int i = tid; i < n; i += 256) sum += __expf(row[i] - m);
  red[tid] = sum; __syncthreads();
  for (int s = 128; s > 0; s >>= 1) { if (tid < s) red[tid] += red[tid + s]; __syncthreads(); }
  float inv = 1.0f / red[0];
  for (int i = tid; i < n; i += 256) row[i] = __expf(row[i] - m) * inv;
}

// ---- DCM: softmax spatial pool of branch0 into concat buffer ---------------
// x.view(B, m, C, q): flat offset mm*80 + cc*4 + qq ; mask.view(B, m, q).
__global__ void k_dcm_ctx(const float* __restrict__ x0, const float* __restrict__ p,
                          float* __restrict__ cat) {
  int t  = blockIdx.x * blockDim.x + threadIdx.x;   // B*20*4096
  int b  = t / (C0 * MINI);
  int r  = t % (C0 * MINI);
  int cc = r / MINI, mm = r % MINI;
  const float4* xs = (const float4*)(x0 + (size_t)b * 40 * HW0 + (size_t)C0 * HW0);
  float4 xv = xs[mm * C0 + cc];                      // (mm*80 + cc*4)/4
  const float4* pv = (const float4*)(p + (size_t)b * HW0);
  float4 pm = pv[mm];
  cat[((size_t)b * CTOT + cc) * MINI + mm] =
      xv.x * pm.x + xv.y * pm.y + xv.z * pm.z + xv.w * pm.w;
}

__global__ void k_dcm_copy(const float* __restrict__ x1, float* __restrict__ cat) {
  int t = blockIdx.x * blockDim.x + threadIdx.x;    // B*40*4096
  int b = t / (C1 * HW1);
  int r = t % (C1 * HW1);
  int c = r / HW1, mm = r % HW1;
  cat[((size_t)b * CTOT + C0 + c) * MINI + mm] = x1[((size_t)b * 80 + C1 + c) * HW1 + mm];
}

// ---- DCM joint SE: fused dual GEMM via V_WMMA_F32_16X16X4_F32 --------------
// Each wave: 16 spatial positions. GEMM1: X(16x60) @ W1^T(60x7) -> relu(bn);
// transpose via LDS into A-frag layout; GEMM2: (16x7) @ W2^T(7x60) -> sigmoid.
// Weights zero-padded in LDS => branch-free inner loop (B-frags are plain
// ds_loads, no exec-mask toggling around the WMMAs).
// Writes attention IN PLACE over `cat` (per-tile reads precede writes).
__global__ void __launch_bounds__(256)
k_dcm_se_wmma(float* __restrict__ cat,
              const float* __restrict__ w1,   // (7,60)
              const float* __restrict__ bng, const float* __restrict__ bnb,
              const float* __restrict__ w2,   // (60,7)
              const float* __restrict__ b2) { // (60)
  __shared__ float sW1[CTOT][16];     // W1^T padded: [k][o], o>=7 -> 0
  __shared__ float sW2[8][65];        // W2^T padded: [k][c], k>=7 | c>=60 -> 0
  __shared__ float sB2[64];
  __shared__ float sSC[16], sBI[16];  // BN scale/bias, zero-padded
  __shared__ float smid[8][16][17];   // [wave][M][K] transpose staging, padded

  const int tid = threadIdx.x;
  for (int i = tid; i < CTOT * 16; i += 256) {
    int k = i >> 4, o = i & 15;
    sW1[k][o] = (o < MID) ? w1[o * CTOT + k] : 0.f;
  }
  for (int i = tid; i < 8 * 64; i += 256) {
    int k = i >> 6, c = i & 63;
    sW2[k][c] = (k < MID && c < CTOT) ? w2[c * MID + k] : 0.f;
  }
  if (tid < 64) sB2[tid] = (tid < CTOT) ? b2[tid] : 0.f;
  if (tid < 16) {
    sSC[tid] = (tid < MID) ? bnscale(bng[tid]) : 0.f;
    sBI[tid] = (tid < MID) ? bnb[tid] : 0.f;
  }
  __syncthreads();

  const int lane = tid & 31;
  const int wv   = tid >> 5;
  const int hl   = lane & 15;                  // half-lane: M (A) / N (B,D)
  const int hi   = lane >> 4;                  // lane half selects K pair / M half
  const int tile = blockIdx.x * 8 + wv;        // 8192 tiles of 16 positions
  const int b    = tile >> 8;                  // 256 tiles per sample
  const int tb   = (tile & 255) * 16;
  const float* X = cat + (size_t)b * CTOT * MINI;

  const v8f vzero = {0.f, 0.f, 0.f, 0.f, 0.f, 0.f, 0.f, 0.f};
  v8f acc = vzero;
  // GEMM1: K = 60, chunks of 4. A lanes 0-15 hold K{0,1}, lanes 16-31 K{2,3}.
  #pragma unroll
  for (int k0 = 0; k0 < CTOT; k0 += 4) {
    const int kk = k0 + 2 * hi;
    v2f a, bb;
    a.x  = X[(size_t)kk * MINI + tb + hl];
    a.y  = X[(size_t)(kk + 1) * MINI + tb + hl];
    bb.x = sW1[kk][hl];                        // B[K][N=hl] (zero-padded)
    bb.y = sW1[kk + 1][hl];
    acc = __builtin_amdgcn_wmma_f32_16x16x4_f32(false, a, false, bb,
                                                (short)0, acc, false, false);
  }
  // D layout: lane -> column N=hl, rows M = hi*8 + r.  BN+ReLU (padded scale
  // gives exact zeros for o>=7), transpose into A-frag order for GEMM2.
  {
    const float sc = sSC[hl];
    const float bi = sBI[hl];
    #pragma unroll
    for (int r = 0; r < 8; ++r)
      smid[wv][hi * 8 + r][hl] = fmaxf(acc[r] * sc + bi, 0.f);
  }
  __syncthreads();
  v8f o0 = vzero, o1 = vzero, o2 = vzero, o3 = vzero;
  #pragma unroll
  for (int k0 = 0; k0 < 8; k0 += 4) {
    const int kk = k0 + 2 * hi;
    v2f a;
    a.x = smid[wv][hl][kk];
    a.y = smid[wv][hl][kk + 1];
    #pragma unroll
    for (int t = 0; t < 4; ++t) {
      v2f bb;
      bb.x = sW2[kk][t * 16 + hl];
      bb.y = sW2[kk + 1][t * 16 + hl];
      v8f& o = (t == 0) ? o0 : (t == 1) ? o1 : (t == 2) ? o2 : o3;
      o = __builtin_amdgcn_wmma_f32_16x16x4_f32(false, a, false, bb,
                                                (short)0, o, false, false);
    }
  }
  #pragma unroll
  for (int t = 0; t < 4; ++t) {
    const int c = t * 16 + hl;
    if (c < CTOT) {
      const v8f& o = (t == 0) ? o0 : (t == 1) ? o1 : (t == 2) ? o2 : o3;
      const float bias = sB2[c];
      float* dst = cat + ((size_t)b * CTOT + c) * MINI + tb + hi * 8;
      #pragma unroll
      for (int r = 0; r < 8; ++r) dst[r] = sigmoidf(o[r] + bias);
    }
  }
}

// ---- apply DCM attention (branch0 needs nearest x2 upsample) ---------------
__global__ void k_apply0(const float* __restrict__ x0, const float* __restrict__ att,
                         float* __restrict__ y0) {
  int t  = blockIdx.x * blockDim.x + threadIdx.x;   // B*20*16384
  int b  = t / (C0 * HW0);
  int r  = t % (C0 * HW0);
  int c  = r / HW0, hw = r % HW0;
  int h  = hw >> 7, w = hw & 127;
  int mi = (h >> 1) * 64 + (w >> 1);
  y0[t] = x0[((size_t)b * 40 + C0 + c) * HW0 + hw] * att[((size_t)b * CTOT + c) * MINI + mi];
}

__global__ void k_apply1(const float* __restrict__ x1, const float* __restrict__ att,
                         float* __restrict__ y1) {
  int t = blockIdx.x * blockDim.x + threadIdx.x;    // B*40*4096
  int b = t / (C1 * HW1);
  int r = t % (C1 * HW1);
  int c = r / HW1, hw = r % HW1;
  y1[t] = x1[((size_t)b * 80 + C1 + c) * HW1 + hw] * att[((size_t)b * CTOT + C0 + c) * MINI + hw];
}

// ---- global average pool: one block per (b,c) ------------------------------
__global__ void k_avgpool(const float* __restrict__ y, float* __restrict__ g, int HW) {
  __shared__ float red[256];
  const float* p = y + (size_t)blockIdx.x * HW;
  float s = 0.f;
  for (int i = threadIdx.x; i < HW; i += 256) s += p[i];
  red[threadIdx.x] = s; __syncthreads();
  for (int st = 128; st > 0; st >>= 1) { if (threadIdx.x < st) red[threadIdx.x] += red[threadIdx.x + st]; __syncthreads(); }
  if (threadIdx.x == 0) g[blockIdx.x] = red[0] / (float)HW;
}

// ---- KernelAttention + kernel aggregation (tiny, one block per sample) -----
__global__ void k_kernel_mix(const float* __restrict__ g,
                             const float* __restrict__ w1, const float* __restrict__ bng,
                             const float* __restrict__ bnb,
                             const float* __restrict__ w2, const float* __restrict__ b2,
                             const float* __restrict__ aggw,  // (4, C, 9)
                             float* __restrict__ mixw,        // (B, C, 9)
                             int C, int mid) {
  __shared__ float sm[16];
  __shared__ float sa[4];
  const int b = blockIdx.x, tid = threadIdx.x;
  const float* gb = g + b * C;
  if (tid < mid) {
    float s = 0.f;
    for (int c = 0; c < C; ++c) s += w1[tid * C + c] * gb[c];
    sm[tid] = fmaxf(s * bnscale(bng[tid]) + bnb[tid], 0.f);
  }
  __syncthreads();
  if (tid < 4) {
    float s = b2[tid];
    for (int j = 0; j < mid; ++j) s += w2[tid * mid + j] * sm[j];
    sa[tid] = sigmoidf(s);
  }
  __syncthreads();
  for (int i = tid; i < C * 9; i += blockDim.x) {
    float s = 0.f;
    #pragma unroll
    for (int k = 0; k < 4; ++k) s += sa[k] * aggw[k * C * 9 + i];
    mixw[b * C * 9 + i] = s;
  }
}

// ---- per-sample depthwise 3x3 (pad 1) + BN ---------------------------------
__global__ void k_dwconv3x3_bn(const float* __restrict__ y, const float* __restrict__ mixw,
                               const float* __restrict__ bng, const float* __restrict__ bnb,
                               float* __restrict__ z, int C, int H, int W) {
  int t  = blockIdx.x * blockDim.x + threadIdx.x;   // B*C*H*W
  int HW = H * W;
  int bc = t / HW, hw = t % HW;
  int c  = bc % C;
  int h  = hw / W, w = hw % W;
  const float* yp = y + (size_t)bc * HW;
  const float* k9 = mixw + (size_t)bc * 9;
  float acc = 0.f;
  #pragma unroll
  for (int dy = -1; dy <= 1; ++dy) {
    int hh = h + dy;
    if ((unsigned)hh >= (unsigned)H) continue;
    #pragma unroll
    for (int dx = -1; dx <= 1; ++dx) {
      int ww = w + dx;
      if ((unsigned)ww >= (unsigned)W) continue;
      acc += yp[hh * W + ww] * k9[(dy + 1) * 3 + (dx + 1)];
    }
  }
  z[t] = acc * bnscale(bng[c]) + bnb[c];
}

// ---- GCM: weighted spatial pool (ctx[b,c] = sum_hw z * p) ------------------
__global__ void k_wpool(const float* __restrict__ z, const float* __restrict__ p,
                        float* __restrict__ ctx, int C, int HW) {
  __shared__ float red[256];
  const int bc = blockIdx.x, b = bc / C;
  const float* zp = z + (size_t)bc * HW;
  const float* pp = p + (size_t)b * HW;
  float s = 0.f;
  for (int i = threadIdx.x; i < HW; i += 256) s += zp[i] * pp[i];
  red[threadIdx.x] = s; __syncthreads();
  for (int st = 128; st > 0; st >>= 1) { if (threadIdx.x < st) red[threadIdx.x] += red[threadIdx.x + st]; __syncthreads(); }
  if (threadIdx.x == 0) ctx[bc] = red[0];
}

// ---- GCM SE MLP (tiny, one block per sample) -------------------------------
__global__ void k_gcm_se(const float* __restrict__ ctx,
                         const float* __restrict__ w1, const float* __restrict__ bng,
                         const float* __restrict__ bnb,
                         const float* __restrict__ w2, const float* __restrict__ b2,
                         float* __restrict__ attn, int C, int mid) {
  __shared__ float sm[16];
  const int b = blockIdx.x, tid = threadIdx.x;
  const float* cb = ctx + b * C;
  if (tid < mid) {
    float s = 0.f;
    for (int c = 0; c < C; ++c) s += w1[tid * C + c] * cb[c];
    sm[tid] = fmaxf(s * bnscale(bng[tid]) + bnb[tid], 0.f);
  }
  __syncthreads();
  if (tid < C) {
    float s = b2[tid];
    for (int j = 0; j < mid; ++j) s += w2[tid * mid + j] * sm[j];
    attn[b * C + tid] = sigmoidf(s);
  }
}

// ---- final: BN(z*attn) + channel-shuffle interleave with first half --------
__global__ void k_final(const float* __restrict__ xfirst, int inC,
                        const float* __restrict__ z, const float* __restrict__ attn,
                        const float* __restrict__ bng, const float* __restrict__ bnb,
                        float* __restrict__ out, int C, int HW) {
  int t = blockIdx.x * blockDim.x + threadIdx.x;    // B*C*HW
  int b = t / (C * HW);
  int r = t % (C * HW);
  int c = r / HW, hw = r % HW;
  float xv = xfirst[((size_t)b * inC + c) * HW + hw];
  float zv = z[t] * attn[b * C + c];
  zv = zv * bnscale(bng[c]) + bnb[c];
  size_t ob = (size_t)b * (2 * C) * HW;
  out[ob + (size_t)(2 * c) * HW + hw]     = xv;   // shuffle: even <- first half
  out[ob + (size_t)(2 * c + 1) * HW + hw] = zv;   // shuffle: odd  <- processed
}

// ---------------------------------------------------------------------------
extern "C" void kernel_launch(void* const* d_in, const int* in_sizes, int n_in,
                              void* d_out, int out_size, void* d_ws, size_t ws_size,
                              hipStream_t stream) {
  (void)in_sizes; (void)n_in; (void)out_size; (void)ws_size;
  const float* x0 = (const float*)d_in[0];
  const float* x1 = (const float*)d_in[1];
  // dcm
  const float* d_mw   = (const float*)d_in[2];
  const float* d_mb   = (const float*)d_in[3];
  const float* d_w1   = (const float*)d_in[4];
  const float* d_bng  = (const float*)d_in[5];
  const float* d_bnb  = (const float*)d_in[6];
  const float* d_w2   = (const float*)d_in[7];
  const float* d_b2   = (const float*)d_in[8];
  // conv[0..1]: att_w1, att_bn_g, att_bn_b, att_w2, att_b2, agg_w, bn_g, bn_b
  const float* cv[2][8];
  for (int i = 0; i < 8; ++i) { cv[0][i] = (const float*)d_in[9 + i]; cv[1][i] = (const float*)d_in[17 + i]; }
  // gcm[0..1]: mask_w, mask_b, ca_w1, ca_bn_g, ca_bn_b, ca_w2, ca_b2, bn_g, bn_b
  const float* gc[2][9];
  for (int i = 0; i < 9; ++i) { gc[0][i] = (const float*)d_in[25 + i]; gc[1][i] = (const float*)d_in[34 + i]; }

  float* ws = (float*)d_ws;
  // workspace layout (floats); CAT doubles as ATT (in-place) then as Z1.
  float* P    = ws;                                  //  524288
  float* CAT  = ws + 524288;                         // 7864320 (B,60,4096)
  float* Y0   = ws + 8388608;                        // 10485760
  float* Y1   = ws + 18874368;                       //  5242880
  float* Z0   = ws + 24117248;                       // 10485760
  float* Z1   = CAT;                                 // reuse (attention dead)
  float* SM   = ws + 34603008;                       // small tensors
  float* g0 = SM, *g1 = SM + 640;
  float* mixw0 = SM + 1920, *mixw1 = SM + 7680;
  float* ctx0 = SM + 19200, *ctx1 = SM + 19840;
  float* sat0 = SM + 21120, *sat1 = SM + 21760;

  float* out0 = (float*)d_out;
  float* out1 = out0 + (size_t)B * 40 * HW0;         // +20971520

  const dim3 T(256);

  // ---------------- DCM ----------------
  k_chan_logits<<<dim3(B * HW0 / 256), T, 0, stream>>>(x0 + (size_t)C0 * HW0, d_mw, d_mb, P,
                                                       C0, HW0, 40 * HW0);
  k_softmax_rows<<<dim3(B), T, 0, stream>>>(P, HW0);
  k_dcm_ctx<<<dim3(B * C0 * MINI / 256), T, 0, stream>>>(x0, P, CAT);
  k_dcm_copy<<<dim3(B * C1 * HW1 / 256), T, 0, stream>>>(x1, CAT);
  k_dcm_se_wmma<<<dim3(B * MINI / (16 * 8)), T, 0, stream>>>(CAT, d_w1, d_bng, d_bnb, d_w2, d_b2);
  k_apply0<<<dim3(B * C0 * HW0 / 256), T, 0, stream>>>(x0, CAT, Y0);
  k_apply1<<<dim3(B * C1 * HW1 / 256), T, 0, stream>>>(x1, CAT, Y1);

  // ---------------- dynamic depthwise conv, branch 0 ----------------
  k_avgpool<<<dim3(B * C0), T, 0, stream>>>(Y0, g0, HW0);
  k_kernel_mix<<<dim3(B), T, 0, stream>>>(g0, cv[0][0], cv[0][1], cv[0][2], cv[0][3], cv[0][4],
                                          cv[0][5], mixw0, C0, C0 / 4);
  k_dwconv3x3_bn<<<dim3(B * C0 * HW0 / 256), T, 0, stream>>>(Y0, mixw0, cv[0][6], cv[0][7],
                                                             Z0, C0, H0, W0);
  // ---------------- dynamic depthwise conv, branch 1 ----------------
  k_avgpool<<<dim3(B * C1), T, 0, stream>>>(Y1, g1, HW1);
  k_kernel_mix<<<dim3(B), T, 0, stream>>>(g1, cv[1][0], cv[1][1], cv[1][2], cv[1][3], cv[1][4],
                                          cv[1][5], mixw1, C1, C1 / 4);
  k_dwconv3x3_bn<<<dim3(B * C1 * HW1 / 256), T, 0, stream>>>(Y1, mixw1, cv[1][6], cv[1][7],
                                                             Z1, C1, H1, W1p);

  // ---------------- GCM + final, branch 0 ----------------
  k_chan_logits<<<dim3(B * HW0 / 256), T, 0, stream>>>(Z0, gc[0][0], gc[0][1], P, C0, HW0, C0 * HW0);
  k_softmax_rows<<<dim3(B), T, 0, stream>>>(P, HW0);
  k_wpool<<<dim3(B * C0), T, 0, stream>>>(Z0, P, ctx0, C0, HW0);
  k_gcm_se<<<dim3(B), dim3(64), 0, stream>>>(ctx0, gc[0][2], gc[0][3], gc[0][4], gc[0][5], gc[0][6],
                                             sat0, C0, C0 / 4);
  k_final<<<dim3(B * C0 * HW0 / 256), T, 0, stream>>>(x0, 40, Z0, sat0, gc[0][7], gc[0][8],
                                                      out0, C0, HW0);
  // ---------------- GCM + final, branch 1 ----------------
  k_chan_logits<<<dim3(B * HW1 / 256), T, 0, stream>>>(Z1, gc[1][0], gc[1][1], P, C1, HW1, C1 * HW1);
  k_softmax_rows<<<dim3(B), T, 0, stream>>>(P, HW1);
  k_wpool<<<dim3(B * C1), T, 0, stream>>>(Z1, P, ctx1, C1, HW1);
  k_gcm_se<<<dim3(B), dim3(64), 0, stream>>>(ctx1, gc[1][2], gc[1][3], gc[1][4], gc[1][5], gc[1][6],
                                             sat1, C1, C1 / 4);
  k_final<<<dim3(B * C1 * HW1 / 256), T, 0, stream>>>(x1, 80, Z1, sat1, gc[1][7], gc[1][8],
                                                      out1, C1, HW1);
}